// BondOrderConv_64407329571242
// MI455X (gfx1250) — compile-verified
//
#include <hip/hip_runtime.h>
#include <math.h>

typedef __attribute__((ext_vector_type(2))) float v2f;
typedef __attribute__((ext_vector_type(8))) float v8f;

#define FEAT 256

// ---------------------------------------------------------------------------
// Kernel 1: per-node gate projections  e_src[n] = <node_feats[n], W_src> + b
//           (and same for W_dst).  One wave (32 lanes) per node; each lane
//           handles 8 features as two coalesced float4 loads, then a wave32
//           shuffle-xor reduction.
// ---------------------------------------------------------------------------
__global__ __launch_bounds__(256) void bond_node_gates(
    const float* __restrict__ node_feats,
    const float* __restrict__ W_src, const float* __restrict__ b_src,
    const float* __restrict__ W_dst, const float* __restrict__ b_dst,
    float* __restrict__ e_src, float* __restrict__ e_dst, int n_nodes)
{
    const int lane = threadIdx.x & 31;
    const int wave = threadIdx.x >> 5;
    const int node = blockIdx.x * 8 + wave;
    if (node >= n_nodes) return;

    const float* row = node_feats + (size_t)node * FEAT;
    float4 x0 = *(const float4*)(row + lane * 4);
    float4 x1 = *(const float4*)(row + lane * 4 + 128);
    float4 s0 = *(const float4*)(W_src + lane * 4);
    float4 s1 = *(const float4*)(W_src + lane * 4 + 128);
    float4 d0 = *(const float4*)(W_dst + lane * 4);
    float4 d1 = *(const float4*)(W_dst + lane * 4 + 128);

    float as = x0.x*s0.x + x0.y*s0.y + x0.z*s0.z + x0.w*s0.w
             + x1.x*s1.x + x1.y*s1.y + x1.z*s1.z + x1.w*s1.w;
    float ad = x0.x*d0.x + x0.y*d0.y + x0.z*d0.z + x0.w*d0.w
             + x1.x*d1.x + x1.y*d1.y + x1.z*d1.z + x1.w*d1.w;

    #pragma unroll
    for (int off = 16; off > 0; off >>= 1) {
        as += __shfl_xor(as, off, 32);
        ad += __shfl_xor(ad, off, 32);
    }
    if (lane == 0) {
        e_src[node] = as + b_src[0];
        e_dst[node] = ad + b_dst[0];
    }
}

// ---------------------------------------------------------------------------
// Kernel 2: edge stream.  Per wave: 16 edges.  WMMA f32 16x16x4 accumulates
// <edge_row, W_edge> over 64 K-chunks.  A: lane L supplies row L%16, K-pair
// 2*(L/16) as one float2 load.  B: the matching W_edge pair broadcast to all
// 16 columns (from LDS).  After the loop, lanes 0-15 hold dots for rows 0-7
// in c[0..7] (all columns identical), lanes 16-31 hold rows 8-15.
// ---------------------------------------------------------------------------
__global__ __launch_bounds__(256) void bond_edge_gate(
    const float* __restrict__ edge_feats,
    const int*   __restrict__ src, const int* __restrict__ dst,
    const float* __restrict__ W_edge, const float* __restrict__ b_edge,
    const float* __restrict__ e_src, const float* __restrict__ e_dst,
    float* __restrict__ out, int n_edges)
{
    __shared__ float w_lds[FEAT];
    const int t = threadIdx.x;
    w_lds[t] = W_edge[t];          // blockDim.x == FEAT == 256
    __syncthreads();

    const int  lane  = t & 31;
    const int  wave  = t >> 5;
    const long long ebase = ((long long)blockIdx.x * 8 + wave) * 16;
    if (ebase >= n_edges) return;
    const float bias = b_edge[0];

    if (ebase + 16 <= (long long)n_edges) {
        const int row = lane & 15;
        const int kh  = (lane >> 4) << 1;            // 0 or 2: which K pair
        const float* arow = edge_feats + (ebase + row) * (long long)FEAT + kh;

        v8f c = {};
        #pragma unroll 8
        for (int k = 0; k < FEAT; k += 4) {
            v2f a = *(const v2f*)(arow + k);
            v2f b = *(const v2f*)(w_lds + k + kh);   // all 16 columns identical
            // (neg_a, A, neg_b, B, c_mod, C, reuse_a, reuse_b)
            c = __builtin_amdgcn_wmma_f32_16x16x4_f32(
                    false, a, false, b, (short)0, c, false, false);
        }

        const int sel = lane & 7;
        float dot;
        switch (sel) {
            case 0: dot = c[0]; break;
            case 1: dot = c[1]; break;
            case 2: dot = c[2]; break;
            case 3: dot = c[3]; break;
            case 4: dot = c[4]; break;
            case 5: dot = c[5]; break;
            case 6: dot = c[6]; break;
            default: dot = c[7]; break;
        }
        // lanes 0-7 own edges 0-7; lanes 16-23 own edges 8-15
        if ((lane & 8) == 0) {
            long long e = ebase + ((lane & 16) ? 8 : 0) + sel;
            float m = dot + e_src[src[e]] + e_dst[dst[e]] + bias;
            out[e] = 1.0f / (1.0f + __expf(-m));
        }
    } else {
        // scalar tail (wave-uniform branch; unused when n_edges % 16 == 0)
        long long e = ebase + lane;
        if (e < n_edges) {
            const float* arow = edge_feats + e * (long long)FEAT;
            float acc = 0.0f;
            for (int k = 0; k < FEAT; ++k) acc += arow[k] * w_lds[k];
            float m = acc + e_src[src[e]] + e_dst[dst[e]] + bias;
            out[e] = 1.0f / (1.0f + __expf(-m));
        }
    }
}

// ---------------------------------------------------------------------------
// Host launcher
// ---------------------------------------------------------------------------
extern "C" void kernel_launch(void* const* d_in, const int* in_sizes, int n_in,
                              void* d_out, int out_size, void* d_ws, size_t ws_size,
                              hipStream_t stream) {
    const float* node_feats = (const float*)d_in[0];
    const float* edge_feats = (const float*)d_in[1];
    const int*   src        = (const int*)d_in[2];
    const int*   dst        = (const int*)d_in[3];
    const float* W_src      = (const float*)d_in[4];
    const float* b_src      = (const float*)d_in[5];
    const float* W_dst      = (const float*)d_in[6];
    const float* b_dst      = (const float*)d_in[7];
    const float* W_edge     = (const float*)d_in[8];
    const float* b_edge     = (const float*)d_in[9];
    float* out = (float*)d_out;

    const int n_nodes = in_sizes[0] / FEAT;   // 10000
    const int n_edges = in_sizes[2];          // 320000

    float* e_src = (float*)d_ws;              // n_nodes floats
    float* e_dst = e_src + n_nodes;           // n_nodes floats (80 KB total)

    const int node_blocks = (n_nodes + 7) / 8;            // 8 nodes / block
    bond_node_gates<<<node_blocks, 256, 0, stream>>>(
        node_feats, W_src, b_src, W_dst, b_dst, e_src, e_dst, n_nodes);

    const int edge_blocks = (int)((n_edges + 127) / 128); // 128 edges / block
    bond_edge_gate<<<edge_blocks, 256, 0, stream>>>(
        edge_feats, src, dst, W_edge, b_edge, e_src, e_dst, out, n_edges);
}